// GTMSM_10179072491645
// MI455X (gfx1250) — compile-verified
//
#include <hip/hip_runtime.h>
#include <hip/hip_bf16.h>
#include <math.h>

typedef __attribute__((ext_vector_type(2))) float v2f;
typedef __attribute__((ext_vector_type(8))) float v8f;

#define Bn 512
#define Tn 150
#define TOn 100
#define TPn 50
#define Zn 8
#define Mn 100
#define Kn 5
#define LOG2PI_F 1.8378770664093453f
#define R_STD_F 0.001f
#define DELTA_F 0.0001f

__device__ __forceinline__ float lrelu(float v) { return v > 0.f ? v : 0.01f * v; }

// ---------------- Kernel 1: state cumsums (one thread per b) ----------------
__global__ void k1_state(const int* __restrict__ action_idx,
                         const float* __restrict__ noise_st,
                         const float* __restrict__ A,
                         float* __restrict__ st) {
  int b = blockIdx.x * blockDim.x + threadIdx.x;
  if (b >= Bn) return;
  float sx = 0.f, sy = 0.f;
  st[(b * Tn + 0) * 2 + 0] = 0.f;
  st[(b * Tn + 0) * 2 + 1] = 0.f;
  for (int t = 0; t < TOn - 1; ++t) {
    int a = action_idx[b * Tn + t];
    sx += A[a * 2 + 0] + R_STD_F * noise_st[(t * Bn + b) * 2 + 0];
    sy += A[a * 2 + 1] + R_STD_F * noise_st[(t * Bn + b) * 2 + 1];
    st[(b * Tn + t + 1) * 2 + 0] = sx;
    st[(b * Tn + t + 1) * 2 + 1] = sy;
  }
  int a99 = action_idx[b * Tn + (TOn - 1)];
  float rx = A[a99 * 2 + 0], ry = A[a99 * 2 + 1];
  float bx = sx, by = sy;
  float nx = 0.f, ny = 0.f;
  for (int i = 1; i <= TPn; ++i) {
    nx += noise_st[((TOn - 1 + i) * Bn + b) * 2 + 0];
    ny += noise_st[((TOn - 1 + i) * Bn + b) * 2 + 1];
    st[(b * Tn + TOn - 1 + i) * 2 + 0] = bx + (float)i * rx + R_STD_F * nx;
    st[(b * Tn + TOn - 1 + i) * 2 + 1] = by + (float)i * ry + R_STD_F * ny;
  }
}

// ---------------- Kernel 2: patch extract + encoder; head GEMM via WMMA ----
// 300 blocks x 256 threads, one sample per thread (B*T = 76800, exact fit).
__global__ void k2_encode(const float* __restrict__ x,
                          const int* __restrict__ position,
                          const float* __restrict__ W1, const float* __restrict__ b1,
                          const float* __restrict__ W2, const float* __restrict__ b2,
                          const float* __restrict__ Wm, const float* __restrict__ bm,
                          const float* __restrict__ Ws, const float* __restrict__ bs,
                          float* __restrict__ meanA, float* __restrict__ stdA,
                          float* __restrict__ patches) {
  __shared__ float fsh[256][33];  // padded to dodge bank conflicts
  const int tid = threadIdx.x;
  const int sid = blockIdx.x * 256 + tid;
  const int b = sid / Tn, t = sid % Tn;

  const int p0 = position[(b * 2 + 0) * Tn + t];
  const int p1 = position[(b * 2 + 1) * Tn + t];
  const float* xb = x + (size_t)b * 3 * 64;

  float pre[3][4];
  #pragma unroll
  for (int c = 0; c < 3; ++c)
    #pragma unroll
    for (int i = 0; i < 2; ++i)
      #pragma unroll
      for (int j = 0; j < 2; ++j) {
        float v = xb[c * 64 + (3 * p0 + i) * 8 + (3 * p1 + j)];
        patches[(size_t)sid * 12 + c * 4 + i * 2 + j] = v;
        pre[c][i * 2 + j] = v * 2.f - 1.f;
      }

  float h1[16][4];
  #pragma unroll
  for (int o = 0; o < 16; ++o) {
    float bo = b1[o];
    #pragma unroll
    for (int p = 0; p < 4; ++p) {
      float s = bo;
      #pragma unroll
      for (int c = 0; c < 3; ++c) s += pre[c][p] * W1[c * 16 + o];
      h1[o][p] = lrelu(s);
    }
  }
  #pragma unroll
  for (int o = 0; o < 8; ++o) {
    float bo = b2[o];
    #pragma unroll
    for (int p = 0; p < 4; ++p) {
      float s = bo;
      #pragma unroll
      for (int c = 0; c < 16; ++c) s += h1[c][p] * W2[c * 8 + o];
      fsh[tid][o * 4 + p] = lrelu(s);  // f = flatten(h2) row-major (c,i,j)
    }
  }
  __syncthreads();

  // ---- WMMA head: out(16 samples x 16 cols) = f(16x32) @ [Wm|Ws](32x16) ----
  const int lane = tid & 31;
  const int wave = tid >> 5;
  const int jcol = lane & 15;        // N index (col 0..7 -> mean, 8..15 -> std)
  const int hi   = lane >> 4;        // lanes 16..31 carry K+2 / M+8 halves

  // Preload B fragments for all 8 K-steps (reused by both 16-row tiles).
  float bf0[8], bf1[8];
  #pragma unroll
  for (int kk = 0; kk < 8; ++kk) {
    int k0 = 4 * kk + hi * 2;
    bf0[kk] = (jcol < 8) ? Wm[k0 * Zn + jcol] : Ws[k0 * Zn + (jcol - 8)];
    bf1[kk] = (jcol < 8) ? Wm[(k0 + 1) * Zn + jcol] : Ws[(k0 + 1) * Zn + (jcol - 8)];
  }
  const float bias = (jcol < 8) ? bm[jcol] : bs[jcol - 8];

  #pragma unroll
  for (int tile = 0; tile < 2; ++tile) {
    const int sbase = wave * 32 + tile * 16;   // first sample row of this tile
    const int arow  = sbase + jcol;            // A-fragment row this lane feeds
    v8f acc;
    #pragma unroll
    for (int r = 0; r < 8; ++r) acc[r] = bias;
    #pragma unroll
    for (int kk = 0; kk < 8; ++kk) {
      int k0 = 4 * kk + hi * 2;
      v2f av, bv;
      av[0] = fsh[arow][k0];
      av[1] = fsh[arow][k0 + 1];
      bv[0] = bf0[kk];
      bv[1] = bf1[kk];
      acc = __builtin_amdgcn_wmma_f32_16x16x4_f32(false, av, false, bv,
                                                  (short)0, acc, false, false);
    }
    #pragma unroll
    for (int r = 0; r < 8; ++r) {
      int srow = sbase + r + hi * 8;           // D: vgpr r -> M=r (lo) / r+8 (hi)
      int gs = blockIdx.x * 256 + srow;
      float v = acc[r];
      if (jcol < 8) meanA[(size_t)gs * Zn + jcol] = v;
      else          stdA[(size_t)gs * Zn + (jcol - 8)] = expf(v);
    }
  }
}

// ---------------- Kernel 3: decode + NLL partials ----------------
__global__ void k3_decode_nll(const float* __restrict__ eps_pred,
                              const float* __restrict__ dW1, const float* __restrict__ db1,
                              const float* __restrict__ dW2, const float* __restrict__ db2,
                              const float* __restrict__ dW3, const float* __restrict__ db3,
                              const float* __restrict__ meanA, const float* __restrict__ stdA,
                              const float* __restrict__ patches,
                              float* __restrict__ nll_part) {
  const int tid = threadIdx.x;
  const int id = blockIdx.x * 256 + tid;     // 25600 total, exact fit
  const int b = id / TPn, tp = id % TPn;
  const int t = TOn + tp;

  float z[8];
  #pragma unroll
  for (int zi = 0; zi < 8; ++zi)
    z[zi] = meanA[((size_t)b * Tn + t) * Zn + zi] +
            stdA[((size_t)b * Tn + t) * Zn + zi] *
            eps_pred[((size_t)b * TPn + tp) * Zn + zi];

  float h1[32];
  #pragma unroll
  for (int k = 0; k < 32; ++k) {
    float s = db1[k];
    #pragma unroll
    for (int zi = 0; zi < 8; ++zi) s += z[zi] * dW1[zi * 32 + k];
    h1[k] = s > 0.f ? s : 0.f;
  }
  float h2[16][4];
  #pragma unroll
  for (int o = 0; o < 16; ++o) {
    float bo = db2[o];
    #pragma unroll
    for (int p = 0; p < 4; ++p) {
      float s = bo;
      #pragma unroll
      for (int c = 0; c < 8; ++c) s += h1[c * 4 + p] * dW2[c * 16 + o];
      h2[o][p] = s > 0.f ? s : 0.f;
    }
  }
  float err = 0.f;
  #pragma unroll
  for (int o = 0; o < 3; ++o) {
    #pragma unroll
    for (int p = 0; p < 4; ++p) {
      float s = db3[o];
      #pragma unroll
      for (int c = 0; c < 16; ++c) s += h2[c][p] * dW3[c * 3 + o];
      float rec = (tanhf(s) + 1.f) * 0.5f;
      float d = rec - patches[((size_t)b * Tn + t) * 12 + o * 4 + p];
      err += d * d;
    }
  }
  __shared__ float red[256];
  red[tid] = 0.5f * err;
  __syncthreads();
  for (int s = 128; s > 0; s >>= 1) {
    if (tid < s) red[tid] += red[tid + s];
    __syncthreads();
  }
  if (tid == 0) nll_part[blockIdx.x] = red[0];
}

// ---------------- Kernel 4: KNN + mixture-param precompute (block per b) ----
__global__ void k4_knn(const float* __restrict__ st,
                       const float* __restrict__ meanA, const float* __restrict__ stdA,
                       float* __restrict__ knnmu, float* __restrict__ knnisd,
                       float* __restrict__ cterm, float* __restrict__ logw,
                       float* __restrict__ logq) {
  const int b = blockIdx.x;
  const int tid = threadIdx.x;  // 64 threads
  __shared__ float so[TOn][2];
  for (int j = tid; j < TOn; j += 64) {
    so[j][0] = st[((size_t)b * Tn + j) * 2 + 0];
    so[j][1] = st[((size_t)b * Tn + j) * 2 + 1];
  }
  __syncthreads();
  if (tid >= TPn) return;
  const int t = tid;
  const float px = st[((size_t)b * Tn + TOn + t) * 2 + 0];
  const float py = st[((size_t)b * Tn + TOn + t) * 2 + 1];

  int chosen[Kn];
  float cd[Kn];
  #pragma unroll
  for (int k = 0; k < Kn; ++k) {
    float best = 3.4e38f;
    int bi = 0;
    for (int j = 0; j < TOn; ++j) {
      bool skip = false;
      for (int q = 0; q < k; ++q) skip = skip || (chosen[q] == j);
      if (skip) continue;
      float dx = px - so[j][0], dy = py - so[j][1];
      float d = dx * dx + dy * dy;
      if (d < best) { best = d; bi = j; }
    }
    chosen[k] = bi;
    cd[k] = best;
  }
  float wk[Kn], ssum = 0.f;
  #pragma unroll
  for (int k = 0; k < Kn; ++k) { wk[k] = 1.f / (cd[k] + DELTA_F); ssum += wk[k]; }
  const size_t base = (size_t)b * TPn + t;
  #pragma unroll
  for (int k = 0; k < Kn; ++k) logw[base * Kn + k] = logf(wk[k] / ssum);

  #pragma unroll
  for (int k = 0; k < Kn; ++k) {
    int j = chosen[k];
    float ct = 0.f;
    #pragma unroll
    for (int zi = 0; zi < Zn; ++zi) {
      float sd = stdA[((size_t)b * Tn + j) * Zn + zi];
      knnmu[(base * Kn + k) * Zn + zi] = meanA[((size_t)b * Tn + j) * Zn + zi];
      knnisd[(base * Kn + k) * Zn + zi] = 1.f / sd;
      ct += -0.5f * LOG2PI_F - logf(sd);
    }
    cterm[base * Kn + k] = ct;
  }
  float lq = -0.5f * (float)Zn * LOG2PI_F - 0.5f * (float)Zn;
  #pragma unroll
  for (int zi = 0; zi < Zn; ++zi)
    lq -= logf(stdA[((size_t)b * Tn + TOn + t) * Zn + zi]);
  logq[base] = lq;
}

// ---------------- Kernel 5: mixture logsumexp sweep (block per (b,t)) ------
__global__ void k5_lp(const float* __restrict__ eps_cluster,
                      const float* __restrict__ meanA, const float* __restrict__ stdA,
                      const float* __restrict__ knnmu, const float* __restrict__ knnisd,
                      const float* __restrict__ cterm, const float* __restrict__ logw,
                      const float* __restrict__ logq, float* __restrict__ kldc) {
  const int bt = blockIdx.x;                // 25600
  const int b = bt / TPn, t = bt % TPn;
  const int tid = threadIdx.x;              // 128
  __shared__ float mu[Kn * Zn], isd[Kn * Zn], ct5[Kn], lw5[Kn], mp[Zn], sp[Zn];
  const size_t base = (size_t)b * TPn + t;
  if (tid < 40) {
    mu[tid] = knnmu[base * 40 + tid];
    isd[tid] = knnisd[base * 40 + tid];
  } else if (tid < 45) {
    ct5[tid - 40] = cterm[base * Kn + (tid - 40)];
  } else if (tid < 50) {
    lw5[tid - 45] = logw[base * Kn + (tid - 45)];
  } else if (tid < 58) {
    mp[tid - 50] = meanA[((size_t)b * Tn + TOn + t) * Zn + (tid - 50)];
    sp[tid - 50] = stdA[((size_t)b * Tn + TOn + t) * Zn + (tid - 50)];
  }
  __syncthreads();

  float lse = 0.f;
  if (tid < Mn) {
    const int m = tid;
    const float* ep = eps_cluster + (((size_t)b * Mn + m) * TPn + t) * Zn;
    float zs[Zn];
    #pragma unroll
    for (int zi = 0; zi < Zn; ++zi) zs[zi] = mp[zi] + sp[zi] * ep[zi];
    float a[Kn];
    #pragma unroll
    for (int k = 0; k < Kn; ++k) {
      float q = 0.f;
      #pragma unroll
      for (int zi = 0; zi < Zn; ++zi) {
        float d = (zs[zi] - mu[k * Zn + zi]) * isd[k * Zn + zi];
        q += d * d;
      }
      a[k] = ct5[k] - 0.5f * q + lw5[k];
    }
    float amax = a[0];
    #pragma unroll
    for (int k = 1; k < Kn; ++k) amax = fmaxf(amax, a[k]);
    float se = 0.f;
    #pragma unroll
    for (int k = 0; k < Kn; ++k) se += expf(a[k] - amax);
    lse = amax + logf(se);
  }
  __shared__ float red[128];
  red[tid] = lse;
  __syncthreads();
  for (int s = 64; s > 0; s >>= 1) {
    if (tid < s) red[tid] += red[tid + s];
    __syncthreads();
  }
  if (tid == 0) {
    float lp_mean = red[0] / (float)Mn;
    kldc[bt] = (logq[base] - lp_mean) / (float)TPn;
  }
}

// ---------------- Kernel 6: deterministic final reduction ----------------
__global__ void k6_final(const float* __restrict__ nll_part,
                         const float* __restrict__ kldc,
                         float* __restrict__ out) {
  __shared__ float red[256];
  const int tid = threadIdx.x;
  float s = 0.f;
  for (int i = tid; i < 100; i += 256) s += nll_part[i];
  for (int i = tid; i < Bn * TPn; i += 256) s += kldc[i];
  red[tid] = s;
  __syncthreads();
  for (int st = 128; st > 0; st >>= 1) {
    if (tid < st) red[tid] += red[tid + st];
    __syncthreads();
  }
  if (tid == 0) out[0] = red[0];
}

extern "C" void kernel_launch(void* const* d_in, const int* in_sizes, int n_in,
                              void* d_out, int out_size, void* d_ws, size_t ws_size,
                              hipStream_t stream) {
  const float* x           = (const float*)d_in[0];
  const int*   action_idx  = (const int*)d_in[1];
  const int*   position    = (const int*)d_in[2];
  const float* noise_st    = (const float*)d_in[3];
  const float* eps_pred    = (const float*)d_in[4];
  const float* eps_cluster = (const float*)d_in[5];
  const float* W1 = (const float*)d_in[6];
  const float* b1 = (const float*)d_in[7];
  const float* W2 = (const float*)d_in[8];
  const float* b2 = (const float*)d_in[9];
  const float* Wm = (const float*)d_in[10];
  const float* bm = (const float*)d_in[11];
  const float* Ws = (const float*)d_in[12];
  const float* bs = (const float*)d_in[13];
  const float* A  = (const float*)d_in[14];
  const float* dW1 = (const float*)d_in[15];
  const float* db1 = (const float*)d_in[16];
  const float* dW2 = (const float*)d_in[17];
  const float* db2 = (const float*)d_in[18];
  const float* dW3 = (const float*)d_in[19];
  const float* db3 = (const float*)d_in[20];

  float* ws = (float*)d_ws;
  float* st      = ws;                       // 512*150*2      = 153600
  float* meanA   = st + 153600;              // 512*150*8      = 614400
  float* stdA    = meanA + 614400;           // 614400
  float* patches = stdA + 614400;            // 512*150*12     = 921600
  float* knnmu   = patches + 921600;         // 512*50*5*8     = 1024000
  float* knnisd  = knnmu + 1024000;          // 1024000
  float* cterm   = knnisd + 1024000;         // 512*50*5       = 128000
  float* logw    = cterm + 128000;           // 128000
  float* logq    = logw + 128000;            // 512*50         = 25600
  float* kldc    = logq + 25600;             // 25600
  float* nllp    = kldc + 25600;             // 100

  k1_state<<<2, 256, 0, stream>>>(action_idx, noise_st, A, st);
  k2_encode<<<300, 256, 0, stream>>>(x, position, W1, b1, W2, b2, Wm, bm, Ws, bs,
                                     meanA, stdA, patches);
  k3_decode_nll<<<100, 256, 0, stream>>>(eps_pred, dW1, db1, dW2, db2, dW3, db3,
                                         meanA, stdA, patches, nllp);
  k4_knn<<<512, 64, 0, stream>>>(st, meanA, stdA, knnmu, knnisd, cterm, logw, logq);
  k5_lp<<<25600, 128, 0, stream>>>(eps_cluster, meanA, stdA, knnmu, knnisd,
                                   cterm, logw, logq, kldc);
  k6_final<<<1, 256, 0, stream>>>(nllp, kldc, (float*)d_out);
}